// PointCloudTVLoss_13958643712287
// MI455X (gfx1250) — compile-verified
//
#include <hip/hip_runtime.h>
#include <math.h>

// ---------------------------------------------------------------------------
// PointCloud TV loss for MI455X (gfx1250, wave32).
//
//   key[c][q] = |p_c|^2 - 2 p_c . p_q   computed with V_WMMA_F32_16X16X4_F32
//   (A row c = (-2x,-2y,-2z,|p_c|^2), B col q = (x,y,z,1), C = 0)
//   d2 = key + |p_q|^2  -> top-16 selection works directly on key.
//   loss = sum_q sum_{16 smallest} sqrt(max(d2,0)+1e-12) / (B*N)
//
// Pipeline: the ds_load of tile t+1's A operand is issued before the top-k
// consumption of tile t's WMMA result, hiding LDS latency behind VALU work.
// ---------------------------------------------------------------------------

typedef __attribute__((ext_vector_type(2))) float v2f;
typedef __attribute__((ext_vector_type(8))) float v8f;

#define NPTS            8192
#define CHUNK           2048                 // candidates staged in LDS per pass (32 KB)
#define TILES_PER_CHUNK (CHUNK / 16)
#define NCHUNKS         (NPTS / CHUNK)
#define WAVES_PER_BLOCK 8
#define BLOCK_THREADS   (WAVES_PER_BLOCK * 32)
#define KNN             16

__global__ __launch_bounds__(BLOCK_THREADS)
void pc_tv_loss_kernel(const float* __restrict__ pc, float* __restrict__ out,
                       float scale)
{
    // Packed candidate tiles: per point (-2x, -2y, -2z, |p|^2)
    __shared__ float lds[CHUNK * 4];

    const int lane = threadIdx.x & 31;
    const int wave = threadIdx.x >> 5;
    const int tid  = threadIdx.x;

    const int tilesPerBatch  = NPTS / 16;                         // 512
    const int blocksPerBatch = tilesPerBatch / WAVES_PER_BLOCK;   // 64
    const int batch = blockIdx.x / blocksPerBatch;
    const int qtile = (blockIdx.x % blocksPerBatch) * WAVES_PER_BLOCK + wave;

    const float* __restrict__ cloud = pc + (size_t)batch * NPTS * 3;

    // ---- B operand: this wave's 16 query points (loop invariant) ----------
    // B 4x16 layout: lane = N (query); lanes 0-15: VGPR0=K0, VGPR1=K1;
    // lanes 16-31: VGPR0=K2, VGPR1=K3.
    const int   q   = qtile * 16 + (lane & 15);
    const float qx  = cloud[3 * q + 0];
    const float qy  = cloud[3 * q + 1];
    const float qz  = cloud[3 * q + 2];
    const float sqq = qx * qx + qy * qy + qz * qz;
    v2f bop;
    if (lane < 16) { bop[0] = qx; bop[1] = qy;   }
    else           { bop[0] = qz; bop[1] = 1.0f; }

    // ---- per-lane sorted top-K selection keys (ascending) ----------------
    float topk[KNN];
#pragma unroll
    for (int t = 0; t < KNN; ++t) topk[t] = 3.402823466e+38f;

    const int m0   = lane & 15;                 // A 16x4: lane = M
    const int aoff = (lane < 16) ? 0 : 2;       // low lanes K=0,1; high K=2,3
    const v8f cz   = {};                        // C = 0 (hoisted)

    for (int c = 0; c < NCHUNKS; ++c) {
        __syncthreads();
        // Stage + transform one candidate chunk into LDS (all global data is
        // L2 resident: 393 KB total, so this is cheap every pass).
        const float* __restrict__ src = cloud + (size_t)c * CHUNK * 3;
        for (int i = tid; i < CHUNK; i += BLOCK_THREADS) {
            float x = src[3 * i + 0];
            float y = src[3 * i + 1];
            float z = src[3 * i + 2];
            lds[4 * i + 0] = -2.0f * x;
            lds[4 * i + 1] = -2.0f * y;
            lds[4 * i + 2] = -2.0f * z;
            lds[4 * i + 3] = x * x + y * y + z * z;
        }
        __syncthreads();

        // Prologue: A operand of tile 0.
        v2f a_cur;
        a_cur[0] = lds[4 * m0 + aoff + 0];
        a_cur[1] = lds[4 * m0 + aoff + 1];

#pragma unroll 2
        for (int t = 0; t < TILES_PER_CHUNK; ++t) {
            // Prefetch tile t+1's A operand (wrapped; the extra final load is
            // harmless) BEFORE consuming this tile's WMMA result, so the LDS
            // wait overlaps the top-k VALU work below.
            const int mn = ((t + 1) & (TILES_PER_CHUNK - 1)) * 16 + m0;
            v2f a_nxt;
            a_nxt[0] = lds[4 * mn + aoff + 0];
            a_nxt[1] = lds[4 * mn + aoff + 1];

            // D[c][q] = |p_c|^2 - 2 p_c.p_q   (16x16 f32 accum, C = 0)
            v8f d = __builtin_amdgcn_wmma_f32_16x16x4_f32(
                false, a_cur, false, bop, (short)0, cz, false, false);

            // Lane owns query q = lane&15 and 8 candidate rows of this tile.
#pragma unroll
            for (int j = 0; j < 8; ++j) {
                float key = d[j];
                if (key < topk[KNN - 1]) {          // early-out threshold
#pragma unroll
                    for (int tt = 0; tt < KNN; ++tt) {
                        float lo = fminf(key, topk[tt]);
                        float hi = fmaxf(key, topk[tt]);
                        topk[tt] = lo;
                        key      = hi;
                    }
                }
            }

            a_cur = a_nxt;
        }
    }

    // ---- merge: lane l (rows 0-7 of tiles) with lane l+16 (rows 8-15) ----
#pragma unroll
    for (int t = 0; t < KNN; ++t) {
        float other = __shfl(topk[t], lane ^ 16, 32);
        if (lane < 16) {
            float key = other;
            if (key < topk[KNN - 1]) {
#pragma unroll
                for (int tt = 0; tt < KNN; ++tt) {
                    float lo = fminf(key, topk[tt]);
                    float hi = fmaxf(key, topk[tt]);
                    topk[tt] = lo;
                    key      = hi;
                }
            }
        }
    }

    // ---- final distances + reduction --------------------------------------
    float s = 0.0f;
    if (lane < 16) {
#pragma unroll
        for (int t = 0; t < KNN; ++t)
            s += sqrtf(fmaxf(topk[t] + sqq, 0.0f) + 1e-12f);
    }
#pragma unroll
    for (int off = 16; off >= 1; off >>= 1)
        s += __shfl_xor(s, off, 32);

    if (lane == 0)
        atomicAdd(out, s * scale);   // global_atomic_add_f32
}

extern "C" void kernel_launch(void* const* d_in, const int* in_sizes, int n_in,
                              void* d_out, int out_size, void* d_ws, size_t ws_size,
                              hipStream_t stream)
{
    const float* pc  = (const float*)d_in[0];
    float*       out = (float*)d_out;

    const int B = in_sizes[0] / (NPTS * 3);      // = 4 for the reference shapes

    // Scalar accumulator must start at zero every call (graph-capture safe).
    hipMemsetAsync(d_out, 0, (out_size > 0 ? out_size : 1) * sizeof(float), stream);

    const float scale  = 1.0f / (float)(B * NPTS);
    const int   blocks = (B * (NPTS / 16)) / WAVES_PER_BLOCK;   // 256

    pc_tv_loss_kernel<<<blocks, BLOCK_THREADS, 0, stream>>>(pc, out, scale);
}